// DualGate_GNN_51539607552125
// MI455X (gfx1250) — compile-verified
//
#include <hip/hip_runtime.h>
#include <hip/hip_bf16.h>

// ---------------------------------------------------------------------------
// DualGate GNN on MI455X (gfx1250, wave32).
// Dense GEMMs use V_WMMA_F32_16X16X4_F32 (fp32 A/B, fp32 accum) to match the
// fp32 reference. Edge scatter/gather (the HBM-bound phase) uses wave-per-edge
// with float2 (b64) coalesced loads + float atomics; gates are wave-per-node
// with __shfl_xor reductions.
// ---------------------------------------------------------------------------

typedef float v2f __attribute__((ext_vector_type(2)));
typedef float v8f __attribute__((ext_vector_type(8)));

__device__ __forceinline__ v8f wmma4(v2f a, v2f b, v8f c) {
  // 8 args: (neg_a, A, neg_b, B, c_mod, C, reuse_a, reuse_b)
  return __builtin_amdgcn_wmma_f32_16x16x4_f32(false, a, false, b, (short)0, c,
                                               false, false);
}

// ---------------------------------------------------------------------------
// Degree / normalization kernels
// ---------------------------------------------------------------------------
__global__ void dg_init_node(float* __restrict__ dinv, float* __restrict__ degR,
                             int n) {
  int i = blockIdx.x * blockDim.x + threadIdx.x;
  if (i < n) {
    dinv[i] = 1.0f;      // self-loop contributes 1 to target degree
    degR[i] = 1e-10f;    // eps in reference
  }
}

__global__ void dg_count_deg(const long long* __restrict__ ei,
                             float* __restrict__ dinv, float* __restrict__ degR,
                             int e_total) {
  int e = blockIdx.x * blockDim.x + threadIdx.x;
  if (e < e_total) {
    int r = (int)ei[e];            // source
    int c = (int)ei[e_total + e];  // target
    atomicAdd(&dinv[c], 1.0f);     // degree over targets (gcn_norm)
    atomicAdd(&degR[r], 1.0f);     // out-degree for gamma_smooth
  }
}

__global__ void dg_finalize_dinv(float* __restrict__ dinv, int n) {
  int i = blockIdx.x * blockDim.x + threadIdx.x;
  if (i < n) dinv[i] = rsqrtf(dinv[i]);
}

// ---------------------------------------------------------------------------
// Encoder + skip:  h = relu(x @ enc_w + enc_b),  xs = x @ skip_w
// One wave per 16-row tile; fp32 WMMA 16x16x4, K = 128.
// ---------------------------------------------------------------------------
__global__ __launch_bounds__(32)
void dg_enc_skip_wmma(const float* __restrict__ x, const float* __restrict__ enc_w,
                      const float* __restrict__ enc_b, const float* __restrict__ skip_w,
                      float* __restrict__ h, float* __restrict__ xs, int n) {
  const int lane = threadIdx.x;
  const int rowBase = blockIdx.x * 16;
  const int m  = lane & 15;              // A: M index
  const int kb = (lane >> 4) << 1;       // A/B: K sub-offset (0 or 2)
  const int nn = lane & 15;              // B/C: N index
  const int rowA = min(rowBase + m, n - 1);

  for (int nt = 0; nt < 4; ++nt) {
    const int colBase = nt * 16;
    v8f accH = {};
    v8f accS = {};
    for (int k0 = 0; k0 < 128; k0 += 4) {
      const float* xp = x + (size_t)rowA * 128 + k0 + kb;
      v2f a;  a[0] = xp[0]; a[1] = xp[1];
      v2f bw; bw[0] = enc_w[(size_t)(k0 + kb) * 64 + colBase + nn];
              bw[1] = enc_w[(size_t)(k0 + kb + 1) * 64 + colBase + nn];
      v2f bs; bs[0] = skip_w[(size_t)(k0 + kb) * 64 + colBase + nn];
              bs[1] = skip_w[(size_t)(k0 + kb + 1) * 64 + colBase + nn];
      accH = wmma4(a, bw, accH);
      accS = wmma4(a, bs, accS);
    }
    const int rOff = (lane >> 4) * 8;
    const int c = colBase + nn;
    const float bias = enc_b[c];
#pragma unroll
    for (int v = 0; v < 8; ++v) {
      int r = rowBase + rOff + v;
      if (r < n) {
        h[(size_t)r * 64 + c]  = fmaxf(accH[v] + bias, 0.0f);
        xs[(size_t)r * 64 + c] = accS[v];
      }
    }
  }
}

// ---------------------------------------------------------------------------
// hw = h @ conv_w   (64x64, fp32 WMMA, K = 64)
// ---------------------------------------------------------------------------
__global__ __launch_bounds__(32)
void dg_conv_wmma(const float* __restrict__ h, const float* __restrict__ conv_w,
                  float* __restrict__ hw, int n) {
  const int lane = threadIdx.x;
  const int rowBase = blockIdx.x * 16;
  const int m  = lane & 15;
  const int kb = (lane >> 4) << 1;
  const int nn = lane & 15;
  const int rowA = min(rowBase + m, n - 1);

  for (int nt = 0; nt < 4; ++nt) {
    const int colBase = nt * 16;
    v8f acc = {};
    for (int k0 = 0; k0 < 64; k0 += 4) {
      const float* hp = h + (size_t)rowA * 64 + k0 + kb;
      v2f a;  a[0] = hp[0]; a[1] = hp[1];
      v2f bw; bw[0] = conv_w[(size_t)(k0 + kb) * 64 + colBase + nn];
              bw[1] = conv_w[(size_t)(k0 + kb + 1) * 64 + colBase + nn];
      acc = wmma4(a, bw, acc);
    }
    const int rOff = (lane >> 4) * 8;
    const int c = colBase + nn;
#pragma unroll
    for (int v = 0; v < 8; ++v) {
      int r = rowBase + rOff + v;
      if (r < n) hw[(size_t)r * 64 + c] = acc[v];
    }
  }
}

// ---------------------------------------------------------------------------
// Per-layer reset (vectorized b128): agg = dinv^2 * hw, diffs = 0, gsum = 0
// ---------------------------------------------------------------------------
__global__ void dg_layer_reset(const float* __restrict__ dinv,
                               const float* __restrict__ hw,
                               float* __restrict__ agg,
                               float* __restrict__ diffs,
                               float* __restrict__ gsum, int n) {
  int idx = blockIdx.x * blockDim.x + threadIdx.x;
  int total = n * 16;                       // n*64 floats = n*16 float4s
  if (idx < total) {
    int i = idx >> 4;                       // node index (16 float4 per node)
    float w = dinv[i];
    float w2 = w * w;
    float4 v = ((const float4*)hw)[idx];
    float4 o;
    o.x = w2 * v.x; o.y = w2 * v.y; o.z = w2 * v.z; o.w = w2 * v.w;
    ((float4*)agg)[idx] = o;
  }
  if (idx < n)  diffs[idx] = 0.0f;
  if (idx < 64) gsum[idx]  = 0.0f;
}

// ---------------------------------------------------------------------------
// Column sums of h (for gmean). 64-thread blocks, one atomic per (block,col).
// ---------------------------------------------------------------------------
__global__ __launch_bounds__(64)
void dg_gmean_accum(const float* __restrict__ h, float* __restrict__ gsum, int n) {
  const int c = threadIdx.x;   // 0..63
  const int ROWS = 256;
  int r0 = blockIdx.x * ROWS;
  int r1 = min(r0 + ROWS, n);
  float s = 0.0f;
  for (int r = r0; r < r1; ++r) s += h[(size_t)r * 64 + c];
  atomicAdd(&gsum[c], s);
}

// ---------------------------------------------------------------------------
// Edge scatter: agg[dst] += norm * hw[src]; diffs[src] += ||h[src]-h[dst]||^2.
// One wave32 per edge; lane owns channels (2*lane, 2*lane+1) -> b64 loads.
// ---------------------------------------------------------------------------
__global__ __launch_bounds__(256)
void dg_edge_scatter(const long long* __restrict__ ei,
                     const float* __restrict__ dinv,
                     const float* __restrict__ h,
                     const float* __restrict__ hw,
                     float* __restrict__ agg,
                     float* __restrict__ diffs, int e_total) {
  const int lane = threadIdx.x & 31;
  const int wid = blockIdx.x * (blockDim.x >> 5) + (threadIdx.x >> 5);
  if (wid >= e_total) return;  // wave-uniform

  const int r = (int)ei[wid];
  const int c = (int)ei[e_total + wid];
  const float w = dinv[r] * dinv[c];
  const size_t rb = (size_t)r * 64, cb = (size_t)c * 64;

  __builtin_prefetch(&hw[rb], 0, 3);  // global_prefetch_b8

  float2 hwv = ((const float2*)(hw + rb))[lane];
  atomicAdd(&agg[cb + 2 * lane],     w * hwv.x);
  atomicAdd(&agg[cb + 2 * lane + 1], w * hwv.y);

  float2 hr = ((const float2*)(h + rb))[lane];
  float2 hc = ((const float2*)(h + cb))[lane];
  float d0 = hr.x - hc.x;
  float d1 = hr.y - hc.y;
  float d = d0 * d0 + d1 * d1;
#pragma unroll
  for (int off = 16; off > 0; off >>= 1) d += __shfl_xor(d, off, 32);
  if (lane == 0) atomicAdd(&diffs[r], d);
}

// ---------------------------------------------------------------------------
// Gates + update: one wave per node; lane owns channels (2*lane, 2*lane+1).
//   gs = tanh(diffs/degR); gq = 1 - tanh(sum |h - gmean|^2.5)
//   h  = (h + gs*relu(agg+conv_b) + gq*xs) / (1 + gs + gq)
// ---------------------------------------------------------------------------
__global__ __launch_bounds__(256)
void dg_gates_update(float* __restrict__ h, const float* __restrict__ agg,
                     const float* __restrict__ xs, const float* __restrict__ conv_b,
                     const float* __restrict__ gsum, const float* __restrict__ diffs,
                     const float* __restrict__ degR, int n, float invN) {
  const int lane = threadIdx.x & 31;
  const int i = blockIdx.x * (blockDim.x >> 5) + (threadIdx.x >> 5);
  if (i >= n) return;  // wave-uniform

  const size_t b = (size_t)i * 64;
  float2 hv = ((const float2*)(h + b))[lane];
  float2 gm = ((const float2*)gsum)[lane];
  gm.x *= invN; gm.y *= invN;

  float a0 = fabsf(hv.x - gm.x), a1 = fabsf(hv.y - gm.y);
  // |v|^2.5 = v^2 * sqrt(|v|)
  float d = a0 * a0 * sqrtf(a0) + a1 * a1 * sqrtf(a1);
#pragma unroll
  for (int off = 16; off > 0; off >>= 1) d += __shfl_xor(d, off, 32);

  float gs = tanhf(diffs[i] / degR[i]);
  float gq = 1.0f - tanhf(d);
  float inv_denom = 1.0f / (1.0f + gs + gq);

  float2 av = ((const float2*)(agg + b))[lane];
  float2 bv = ((const float2*)conv_b)[lane];
  float2 sv = ((const float2*)(xs + b))[lane];
  float ag0 = fmaxf(av.x + bv.x, 0.0f);
  float ag1 = fmaxf(av.y + bv.y, 0.0f);

  float2 o;
  o.x = (hv.x + gs * ag0 + gq * sv.x) * inv_denom;
  o.y = (hv.y + gs * ag1 + gq * sv.y) * inv_denom;
  ((float2*)(h + b))[lane] = o;
}

// ---------------------------------------------------------------------------
// Decoder: out = h @ dec_w + dec_b   (64 -> 40; 3rd col-tile partial)
// Branch-free B loads keep EXEC all-ones around the WMMA chain.
// ---------------------------------------------------------------------------
__global__ __launch_bounds__(32)
void dg_dec_wmma(const float* __restrict__ h, const float* __restrict__ dec_w,
                 const float* __restrict__ dec_b, float* __restrict__ out, int n) {
  const int lane = threadIdx.x;
  const int rowBase = blockIdx.x * 16;
  const int m  = lane & 15;
  const int kb = (lane >> 4) << 1;
  const int nn = lane & 15;
  const int rowA = min(rowBase + m, n - 1);

  for (int nt = 0; nt < 3; ++nt) {
    const int colBase = nt * 16;
    const int c = colBase + nn;
    const int cc = c < 40 ? c : 39;      // clamped (in-bounds) address
    const float mask = (c < 40) ? 1.0f : 0.0f;
    v8f acc = {};
    for (int k0 = 0; k0 < 64; k0 += 4) {
      const float* hp = h + (size_t)rowA * 64 + k0 + kb;
      v2f a;  a[0] = hp[0]; a[1] = hp[1];
      v2f bw; bw[0] = dec_w[(size_t)(k0 + kb) * 40 + cc] * mask;
              bw[1] = dec_w[(size_t)(k0 + kb + 1) * 40 + cc] * mask;
      acc = wmma4(a, bw, acc);
    }
    if (c < 40) {
      const float bias = dec_b[c];
      const int rOff = (lane >> 4) * 8;
#pragma unroll
      for (int v = 0; v < 8; ++v) {
        int r = rowBase + rOff + v;
        if (r < n) out[(size_t)r * 40 + c] = acc[v] + bias;
      }
    }
  }
}

// ---------------------------------------------------------------------------
extern "C" void kernel_launch(void* const* d_in, const int* in_sizes, int n_in,
                              void* d_out, int out_size, void* d_ws, size_t ws_size,
                              hipStream_t stream) {
  const float*     x      = (const float*)d_in[0];
  const long long* ei     = (const long long*)d_in[1];  // int64 [2,E]
  const float*     enc_w  = (const float*)d_in[2];
  const float*     enc_b  = (const float*)d_in[3];
  const float*     skip_w = (const float*)d_in[4];
  const float*     conv_w = (const float*)d_in[5];
  const float*     conv_b = (const float*)d_in[6];
  const float*     dec_w  = (const float*)d_in[7];
  const float*     dec_b  = (const float*)d_in[8];

  const int n = in_sizes[0] / 128;   // 100000
  const int e = in_sizes[1] / 2;     // 1200000

  // Workspace layout (floats); large arrays are 16B aligned when n % 4 == 0.
  float* ws    = (float*)d_ws;
  float* dinv  = ws;                       // N   (holds degT then rsqrt)
  float* degR  = dinv + n;                 // N
  float* diffs = degR + n;                 // N
  float* gsum  = diffs + n;                // 64
  float* h     = gsum + 64;                // N*64
  float* xs    = h  + (size_t)n * 64;      // N*64
  float* hw    = xs + (size_t)n * 64;      // N*64
  float* agg   = hw + (size_t)n * 64;      // N*64

  const int mtiles = (n + 15) / 16;

  dg_init_node    <<<(n + 255) / 256, 256, 0, stream>>>(dinv, degR, n);
  dg_count_deg    <<<(e + 255) / 256, 256, 0, stream>>>(ei, dinv, degR, e);
  dg_finalize_dinv<<<(n + 255) / 256, 256, 0, stream>>>(dinv, n);

  dg_enc_skip_wmma<<<mtiles, 32, 0, stream>>>(x, enc_w, enc_b, skip_w, h, xs, n);

  for (int layer = 0; layer < 2; ++layer) {
    dg_conv_wmma   <<<mtiles, 32, 0, stream>>>(h, conv_w, hw, n);
    dg_layer_reset <<<((size_t)n * 16 + 255) / 256, 256, 0, stream>>>(
        dinv, hw, agg, diffs, gsum, n);
    dg_gmean_accum <<<(n + 255) / 256, 64, 0, stream>>>(h, gsum, n);
    dg_edge_scatter<<<(e + 7) / 8, 256, 0, stream>>>(ei, dinv, h, hw, agg, diffs, e);
    dg_gates_update<<<(n + 7) / 8, 256, 0, stream>>>(
        h, agg, xs, conv_b, gsum, diffs, degR, n, 1.0f / (float)n);
  }

  dg_dec_wmma<<<mtiles, 32, 0, stream>>>(h, dec_w, dec_b, (float*)d_out, n);
}